// DAF_55679956025596
// MI455X (gfx1250) — compile-verified
//
#include <hip/hip_runtime.h>
#include <hip/hip_bf16.h>
#include <math.h>

// ---------------------------------------------------------------------------
// DAF forecaster for MI455X (gfx1250), fp32 end-to-end.
// - All MLPs are affine (leaky relu only at the end) -> collapsed on device to
//   single K-padded, transposed (c-major) weight blocks + bias row.
// - All big matmuls use V_WMMA_F32_16X16X4_F32, one wave per 16x16 tile.
// - Loads are unconditional (K padded with zero B rows; overreads stay inside
//   d_ws); only stores are bounds-guarded. Contiguous-k operands use b64 loads.
// - Steps 0..2 only need the rep_ex path for the appended column; the full
//   O(T^2) attention runs only on the final step.
// ---------------------------------------------------------------------------

typedef float v2f __attribute__((ext_vector_type(2)));
typedef float v8f __attribute__((ext_vector_type(8)));

#define NEG_SLOPE 0.01f

// ---------------- generic strided WMMA GEMM: C = leaky?(A@B + bias) --------
// A: logical (M,K), element Ab[r*aR + k*aC], Ab = A + b*aB
// B: logical (K,N), element Bb[k*bR + c*bC]   (K already padded to mult of 4,
//    rows >= real K must be zero in B)
// C: logical (M,N), element Cb[r*cR + c*cC]
__global__ void daf_wmma_gemm_kernel(
    const float* __restrict__ A, long aB, long aR, long aC,
    const float* __restrict__ Bm, long bB, long bR, long bC,
    float* __restrict__ C, long cB, long cR, long cC,
    const float* __restrict__ bias, int M, int N, int K, int doLeaky)
{
    int b   = blockIdx.z;
    int m0  = blockIdx.y * 16;
    int n0  = blockIdx.x * 16;
    int lane = threadIdx.x & 31;
    int r16  = lane & 15;
    int hi   = lane >> 4;          // k pair select: 0 -> {k0,k0+1}, 1 -> {k0+2,k0+3}

    // pre-offset pointers: per-iteration address is ptr + k0*stride
    const float* Ap = A + (long)b*aB + (long)(m0 + r16)*aR + (long)(2*hi)*aC;
    const float* Bp = Bm + (long)b*bB + (long)(n0 + r16)*bC + (long)(2*hi)*bR;

    v8f acc = {0.f,0.f,0.f,0.f,0.f,0.f,0.f,0.f};

    if (aC == 1 && bR == 1) {
#pragma unroll 2
        for (int k0 = 0; k0 < K; k0 += 4) {
            v2f av = *(const v2f*)(Ap + k0);
            v2f bv = *(const v2f*)(Bp + k0);
            acc = __builtin_amdgcn_wmma_f32_16x16x4_f32(
                      false, av, false, bv, (short)0, acc, false, false);
        }
    } else if (aC == 1) {
#pragma unroll 2
        for (int k0 = 0; k0 < K; k0 += 4) {
            v2f av = *(const v2f*)(Ap + k0);
            v2f bv;
            bv.x = Bp[(long)k0*bR];
            bv.y = Bp[(long)k0*bR + bR];
            acc = __builtin_amdgcn_wmma_f32_16x16x4_f32(
                      false, av, false, bv, (short)0, acc, false, false);
        }
    } else if (bR == 1) {
#pragma unroll 2
        for (int k0 = 0; k0 < K; k0 += 4) {
            v2f av;
            av.x = Ap[(long)k0*aC];
            av.y = Ap[(long)k0*aC + aC];
            v2f bv = *(const v2f*)(Bp + k0);
            acc = __builtin_amdgcn_wmma_f32_16x16x4_f32(
                      false, av, false, bv, (short)0, acc, false, false);
        }
    } else {
#pragma unroll 2
        for (int k0 = 0; k0 < K; k0 += 4) {
            v2f av, bv;
            av.x = Ap[(long)k0*aC];
            av.y = Ap[(long)k0*aC + aC];
            bv.x = Bp[(long)k0*bR];
            bv.y = Bp[(long)k0*bR + bR];
            acc = __builtin_amdgcn_wmma_f32_16x16x4_f32(
                      false, av, false, bv, (short)0, acc, false, false);
        }
    }

    float* Cb = C + (long)b * cB;
#pragma unroll
    for (int v = 0; v < 8; ++v) {
        int r = m0 + hi*8 + v;
        int c = n0 + r16;
        if (r < M && c < N) {
            float x = acc[v];
            if (bias)    x += bias[c];
            if (doLeaky) x = x > 0.f ? x : NEG_SLOPE * x;
            Cb[(long)r*cR + (long)c*cC] = x;
        }
    }
}

// -------- collapse stage 1: tmp[i,j] = a(i)@W1 + (last row: +b1) ------------
// a(i,k) = (i==R-1) ? b0[k] : W0[i*Kd+k];  tmp is (R x 64) row-major
__global__ void daf_collapse1_kernel(const float* __restrict__ W0,
                                     const float* __restrict__ b0,
                                     int R, int Kd,
                                     const float* __restrict__ W1,
                                     const float* __restrict__ b1,
                                     int N, float* __restrict__ tmp)
{
    int idx = blockIdx.x * blockDim.x + threadIdx.x;
    if (idx >= R * N) return;
    int i = idx / N, j = idx % N;
    bool brow = (i == R - 1);
    float s = 0.f;
    for (int k = 0; k < Kd; ++k) {
        float a = brow ? b0[k] : W0[i*Kd + k];
        s += a * W1[k*N + j];
    }
    if (brow) s += b1[j];
    tmp[i*N + j] = s;
}

// -------- collapse stage 2: emit transposed + K-padded weights --------------
// out[j*Kpad + i] = (tmp@W2)[i,j] for i<din; 0 for din<=i<Kpad;
// out[N*Kpad + j] = bias_j = tmp[din,:]@W2[:,j] + b2[j]
__global__ void daf_collapse2_kernel(const float* __restrict__ tmp,
                                     int din, int Kpad,
                                     const float* __restrict__ W2,
                                     const float* __restrict__ b2,
                                     int N, float* __restrict__ out)
{
    int idx = blockIdx.x * blockDim.x + threadIdx.x;
    if (idx >= N * (Kpad + 1)) return;
    int j = idx / (Kpad + 1);
    int i = idx % (Kpad + 1);
    if (i == Kpad) {                      // bias row
        float s = b2[j];
        for (int k = 0; k < 64; ++k) s += tmp[din*64 + k] * W2[k*N + j];
        out[(long)N*Kpad + j] = s;
    } else if (i >= din) {                // zero padding rows
        out[(long)j*Kpad + i] = 0.f;
    } else {
        float s = 0.f;
        for (int k = 0; k < 64; ++k) s += tmp[i*64 + k] * W2[k*N + j];
        out[(long)j*Kpad + i] = s;
    }
}

// -------- copy input (B,129,512) into padded (B,129,516) data buffer --------
__global__ void daf_copy_in_kernel(const float* __restrict__ src,
                                   float* __restrict__ data_buf, int total)
{
    int idx = blockIdx.x * blockDim.x + threadIdx.x;
    if (idx >= total) return;
    int b = idx / (129 * 512);
    int r = idx - b * 129 * 512;
    int c = r / 512;
    int t = r - c * 512;
    data_buf[(long)b*129*516 + (long)c*516 + t] = src[idx];
}

// -------- zero rows [rowStart,rowEnd) of a (B, rows, 128) buffer ------------
__global__ void daf_zero_rows_kernel(float* __restrict__ buf, long bstride,
                                     int rowStart, int rowEnd)
{
    int b = blockIdx.x;
    int tid = threadIdx.x;
    for (int r = rowStart; r < rowEnd; ++r)
        buf[(long)b*bstride + (long)r*128 + tid] = 0.f;
}

// -------- conv1: (B,129,T) -> h1 (B,T,64), ks=3 pad=1, bias, no act ---------
__global__ void daf_conv1_kernel(const float* __restrict__ data,
                                 const float* __restrict__ W1,
                                 const float* __restrict__ b1,
                                 float* __restrict__ h1, int T)
{
    int b = blockIdx.y;
    int idx = blockIdx.x * blockDim.x + threadIdx.x;
    if (idx >= T * 64) return;
    int tt = idx >> 6;
    int o  = idx & 63;
    const float* dbase = data + (long)b*129*516;
    float s = b1[o];
    for (int i = 0; i < 129; ++i) {
        const float* drow = dbase + (long)i*516;
        const float* wrow = W1 + ((long)o*129 + i)*3;
#pragma unroll
        for (int k = 0; k < 3; ++k) {
            int st = tt + k - 1;
            if (st >= 0 && st < T) s += wrow[k] * drow[st];
        }
    }
    h1[(long)b*516*64 + (long)tt*64 + o] = s;
}

// -------- conv2: h1 (B,T,64) -> pattern (B,T,128), ks=5 pad=2, bias+leaky ---
__global__ void daf_conv2_kernel(const float* __restrict__ h1,
                                 const float* __restrict__ W2,
                                 const float* __restrict__ b2,
                                 float* __restrict__ pat, int T)
{
    int b = blockIdx.y;
    int idx = blockIdx.x * blockDim.x + threadIdx.x;
    if (idx >= T * 128) return;
    int tt = idx >> 7;
    int o  = idx & 127;
    const float* hb = h1 + (long)b*516*64;
    float s = b2[o];
#pragma unroll
    for (int k = 0; k < 5; ++k) {
        int st = tt + k - 2;
        if (st < 0 || st >= T) continue;
        const float* hrow = hb + (long)st*64;
        const float* wcol = W2 + (long)o*64*5 + k;     // W2[(o*64+i)*5+k]
        for (int i = 0; i < 64; ++i) s += wcol[i*5] * hrow[i];
    }
    s = s > 0.f ? s : NEG_SLOPE * s;
    pat[(long)b*516*128 + (long)tt*128 + o] = s;
}

// -------- in-place A = softmax(exp(S - S*eye/sqrt(T))) per row --------------
__global__ void daf_attn_softmax_kernel(float* __restrict__ S, int T,
                                        long sB, long sR, float invsqrtT)
{
    int b = blockIdx.y;
    int t = blockIdx.x;
    float* row = S + (long)b*sB + (long)t*sR;
    __shared__ float red[256];
    int tid = threadIdx.x;
    float diag = row[t];
    float dadj = diag - diag * invsqrtT;

    float m = -INFINITY;
    for (int s = tid; s < T; s += 256) {
        float v = (s == t) ? dadj : row[s];
        m = fmaxf(m, v);
    }
    red[tid] = m; __syncthreads();
    for (int o = 128; o > 0; o >>= 1) {
        if (tid < o) red[tid] = fmaxf(red[tid], red[tid+o]);
        __syncthreads();
    }
    m = red[0]; __syncthreads();
    float L = expf(m);              // max of the (monotone) exp-logits

    float sum = 0.f;
    for (int s = tid; s < T; s += 256) {
        float v = (s == t) ? dadj : row[s];
        float w = expf(expf(v) - L);
        row[s] = w;
        sum += w;
    }
    red[tid] = sum; __syncthreads();
    for (int o = 128; o > 0; o >>= 1) {
        if (tid < o) red[tid] += red[tid+o];
        __syncthreads();
    }
    float inv = 1.f / red[0];
    for (int s = tid; s < T; s += 256) row[s] *= inv;
}

// -------- ex-path attention (kernel_size=5, Tq=1) -> exbuf (B,128) ----------
// WBq layout: weights at [d*128 + c], bias at [128*128 + d]
__global__ void daf_attn_ex_kernel(const float* __restrict__ pat, long patB,
                                   const float* __restrict__ kb, long kBs,
                                   const float* __restrict__ vb, long vBs,
                                   const float* __restrict__ WBq,
                                   float* __restrict__ exbuf, int T)
{
    int b = blockIdx.x;
    int tid = threadIdx.x;
    __shared__ float q0[128];
    __shared__ float w[512];
    __shared__ float red[256];
    int L = T - 6;
    const float* prow = pat + (long)b*patB + (long)(T-3)*128;
    if (tid < 128) {
        float s = WBq[128*128 + tid];
        const float* wr = WBq + (long)tid*128;
        for (int c = 0; c < 128; ++c) s += prow[c] * wr[c];
        q0[tid] = s > 0.f ? s : NEG_SLOPE * s;
    }
    __syncthreads();
    for (int s = tid; s < L; s += 256) {
        if (s == 0) { w[0] = 1.0f; continue; }   // diag: exp(S - S) = 1
        const float* krow = kb + (long)b*kBs + (long)(2+s)*128;
        float acc = 0.f;
        for (int d = 0; d < 128; ++d) acc += q0[d] * krow[d];
        w[s] = expf(acc);
    }
    __syncthreads();
    float m = -INFINITY;
    for (int s = tid; s < L; s += 256) m = fmaxf(m, w[s]);
    red[tid] = m; __syncthreads();
    for (int o = 128; o > 0; o >>= 1) {
        if (tid < o) red[tid] = fmaxf(red[tid], red[tid+o]);
        __syncthreads();
    }
    m = red[0]; __syncthreads();
    float sum = 0.f;
    for (int s = tid; s < L; s += 256) { float e = expf(w[s]-m); w[s] = e; sum += e; }
    red[tid] = sum; __syncthreads();
    for (int o = 128; o > 0; o >>= 1) {
        if (tid < o) red[tid] += red[tid+o];
        __syncthreads();
    }
    float inv = 1.f / red[0];
    __syncthreads();
    if (tid < 128) {
        float acc = 0.f;
        for (int s = 0; s < L; ++s)
            acc += w[s] * vb[(long)b*vBs + (long)(5+s)*128 + tid];
        exbuf[b*128 + tid] = acc * inv;
    }
}

// -------- steps 0..2 tail: o-mlp + dec-mlp on the single ex row, append ----
// WBo: weights [d2*128 + d], bias [128*128 + d2]
// WBdec: weights [c*128 + d], bias [129*128 + c]
__global__ void daf_ex_tail_kernel(const float* __restrict__ exbuf,
                                   const float* __restrict__ WBo,
                                   const float* __restrict__ WBdec,
                                   float* __restrict__ data_buf,
                                   long dataB, int T)
{
    int b = blockIdx.x;
    int tid = threadIdx.x;
    __shared__ float rep[128];
    if (tid < 128) {
        float s = WBo[128*128 + tid];
        const float* wr = WBo + (long)tid*128;
        for (int d = 0; d < 128; ++d) s += exbuf[b*128 + d] * wr[d];
        rep[tid] = s > 0.f ? s : NEG_SLOPE * s;
    }
    __syncthreads();
    if (tid < 129) {
        float s = WBdec[129*128 + tid];
        const float* wr = WBdec + (long)tid*128;
        for (int d = 0; d < 128; ++d) s += rep[d] * wr[d];
        s = s > 0.f ? s : NEG_SLOPE * s;
        data_buf[(long)b*dataB + (long)tid*516 + T] = s;   // append column T
    }
}

// -------- copy exbuf into row T of the concatenated attention buffer --------
__global__ void daf_copy_ex_row_kernel(const float* __restrict__ exbuf,
                                       float* __restrict__ cat, long catB, int T)
{
    int b = blockIdx.x;
    int d = threadIdx.x;
    cat[(long)b*catB + (long)T*128 + d] = exbuf[b*128 + d];
}

// ---------------------------------------------------------------------------
static inline void launch_gemm(hipStream_t st,
    const float* A, long aB, long aR, long aC,
    const float* Bm, long bB, long bR, long bC,
    float* C, long cB, long cR, long cC,
    const float* bias, int M, int N, int K, int leaky, int batch)
{
    dim3 g((N + 15) / 16, (M + 15) / 16, batch);
    daf_wmma_gemm_kernel<<<g, 32, 0, st>>>(A, aB, aR, aC, Bm, bB, bR, bC,
                                           C, cB, cR, cC, bias, M, N, K, leaky);
}

static inline void collapse_mlp(hipStream_t st,
    const float* W0, const float* b0, const float* W1, const float* b1,
    const float* W2, const float* b2, int din, int dout,
    float* tmp, float* out)
{
    int R = din + 1;
    int Kpad = (din + 3) & ~3;
    int n1 = R * 64;
    daf_collapse1_kernel<<<(n1 + 255) / 256, 256, 0, st>>>(W0, b0, R, 64, W1, b1, 64, tmp);
    int n2 = dout * (Kpad + 1);
    daf_collapse2_kernel<<<(n2 + 255) / 256, 256, 0, st>>>(tmp, din, Kpad, W2, b2, dout, out);
}

extern "C" void kernel_launch(void* const* d_in, const int* in_sizes, int n_in,
                              void* d_out, int out_size, void* d_ws, size_t ws_size,
                              hipStream_t stream)
{
    (void)in_sizes; (void)n_in; (void)out_size; (void)ws_size;
    // Input order (setup_inputs dict insertion order, params flattened):
    //  0 src_data  1 tgt_data
    //  src_enc.conv: 2 W1  3 b1  4 W2  5 b2
    //  src_enc.v:    6 W0  7 W1  8 W2  9 b0 10 b1 11 b2
    //  tgt_enc.conv:12 W1 13 b1 14 W2 15 b2
    //  tgt_enc.v:   16 W0 17 W1 18 W2 19 b0 20 b1 21 b2
    //  attn.q:      22 W0 23 W1 24 W2 25 b0 26 b1 27 b2
    //  attn.k:      28..33      attn.o: 34..39
    //  src_dec:     40..45      tgt_dec: 46..51
    #define FI(i) ((const float*)d_in[(i)])

    const int Bn = 32, Dm = 129, Cc = 128, TMAX = 516, T0 = 512, PRED = 4;
    float* ws = (float*)d_ws;
    long o = 0;
    float* data_buf = ws + o; o += (long)Bn*Dm*TMAX;
    float* h1    = ws + o; o += (long)Bn*TMAX*64;
    float* pat   = ws + o; o += (long)Bn*TMAX*Cc;
    float* val   = ws + o; o += (long)Bn*TMAX*Cc;
    float* kbuf  = ws + o; o += (long)Bn*TMAX*Cc;
    float* qbuf  = ws + o; o += (long)Bn*TMAX*Cc;
    float* Sbuf  = ws + o; o += (long)Bn*TMAX*TMAX;
    float* cat   = ws + o; o += (long)Bn*517*Cc;
    float* rep   = ws + o; o += (long)Bn*517*Cc;
    float* exbuf = ws + o; o += (long)Bn*Cc;
    // collapsed weights, all sizes rounded to multiples of 4 floats (alignment)
    float* WB_vs = ws + o; o += 17024;   // 128*132 + 128
    float* WB_vt = ws + o; o += 17024;
    float* WB_q  = ws + o; o += 16512;   // 128*128 + 128
    float* WB_k  = ws + o; o += 16512;
    float* WB_o  = ws + o; o += 16512;
    float* WB_ds = ws + o; o += 16644;   // 129*128 + 129, rounded up
    float* WB_dt = ws + o; o += 16644;
    float* tmp   = ws + o; o += 130*64;

    // collapse all affine MLPs once per launch (transposed + K-padded)
    collapse_mlp(stream, FI(6),  FI(9),  FI(7),  FI(10), FI(8),  FI(11), 129, 128, tmp, WB_vs);
    collapse_mlp(stream, FI(16), FI(19), FI(17), FI(20), FI(18), FI(21), 129, 128, tmp, WB_vt);
    collapse_mlp(stream, FI(22), FI(25), FI(23), FI(26), FI(24), FI(27), 128, 128, tmp, WB_q);
    collapse_mlp(stream, FI(28), FI(31), FI(29), FI(32), FI(30), FI(33), 128, 128, tmp, WB_k);
    collapse_mlp(stream, FI(34), FI(37), FI(35), FI(38), FI(36), FI(39), 128, 128, tmp, WB_o);
    collapse_mlp(stream, FI(40), FI(43), FI(41), FI(44), FI(42), FI(45), 128, 129, tmp, WB_ds);
    collapse_mlp(stream, FI(46), FI(49), FI(47), FI(50), FI(48), FI(51), 128, 129, tmp, WB_dt);

    for (int br = 0; br < 2; ++br) {
        const float* din0 = FI(br);
        const float* cW1 = FI(br ? 12 : 2);
        const float* cb1 = FI(br ? 13 : 3);
        const float* cW2 = FI(br ? 14 : 4);
        const float* cb2 = FI(br ? 15 : 5);
        const float* WBv = br ? WB_vt : WB_vs;
        const float* WBd = br ? WB_dt : WB_ds;
        float* outb = (float*)d_out + (long)br * Bn * Dm * TMAX;

        int tot = Bn * Dm * T0;
        daf_copy_in_kernel<<<(tot + 255) / 256, 256, 0, stream>>>(din0, data_buf, tot);

        for (int step = 0; step < PRED; ++step) {
            int T = T0 + step;
            daf_conv1_kernel<<<dim3((T*64  + 255)/256, Bn), 256, 0, stream>>>(data_buf, cW1, cb1, h1, T);
            daf_conv2_kernel<<<dim3((T*128 + 255)/256, Bn), 256, 0, stream>>>(h1, cW2, cb2, pat, T);

            // value = leaky(data^T @ Wv + bv): A col-major (aC=516), B b64 path
            launch_gemm(stream, data_buf, (long)Dm*TMAX, 1L, (long)TMAX,
                        WBv, 0, 1, 132,
                        val, (long)TMAX*Cc, Cc, 1,
                        WBv + 128*132, T, 128, 132, 1, Bn);
            // k = leaky(pattern @ Wk + bk): both sides b64
            launch_gemm(stream, pat, (long)TMAX*Cc, Cc, 1,
                        WB_k, 0, 1, 128,
                        kbuf, (long)TMAX*Cc, Cc, 1,
                        WB_k + 128*128, T, 128, 128, 1, Bn);

            if (step < PRED - 1) {
                // only the appended column matters -> ex path only
                daf_attn_ex_kernel<<<Bn, 256, 0, stream>>>(
                    pat, (long)TMAX*Cc, kbuf, (long)TMAX*Cc, val, (long)TMAX*Cc,
                    WB_q, exbuf, T);
                daf_ex_tail_kernel<<<Bn, 160, 0, stream>>>(
                    exbuf, WB_o, WBd, data_buf, (long)Dm*TMAX, T);
            } else {
                int Tpad = (T + 3) & ~3;
                // q = leaky(pattern @ Wq + bq)
                launch_gemm(stream, pat, (long)TMAX*Cc, Cc, 1,
                            WB_q, 0, 1, 128,
                            qbuf, (long)TMAX*Cc, Cc, 1,
                            WB_q + 128*128, T, 128, 128, 1, Bn);
                // S = Q @ K^T (both operands contiguous in k)
                launch_gemm(stream, qbuf, (long)TMAX*Cc, Cc, 1,
                            kbuf, (long)TMAX*Cc, 1, Cc,
                            Sbuf, (long)TMAX*TMAX, TMAX, 1,
                            nullptr, T, T, 128, 0, Bn);
                daf_attn_softmax_kernel<<<dim3(T, Bn), 256, 0, stream>>>(
                    Sbuf, T, (long)TMAX*TMAX, (long)TMAX, 1.0f / sqrtf((float)T));
                // zero val rows [T, Tpad) so the K-padded A@V adds zeros
                if (Tpad > T)
                    daf_zero_rows_kernel<<<Bn, 128, 0, stream>>>(val, (long)TMAX*Cc, T, Tpad);
                // cat[0..T-1] = A @ V
                launch_gemm(stream, Sbuf, (long)TMAX*TMAX, TMAX, 1,
                            val, (long)TMAX*Cc, Cc, 1,
                            cat, 517L*Cc, Cc, 1,
                            nullptr, T, 128, Tpad, 0, Bn);
                daf_attn_ex_kernel<<<Bn, 256, 0, stream>>>(
                    pat, (long)TMAX*Cc, kbuf, (long)TMAX*Cc, val, (long)TMAX*Cc,
                    WB_q, exbuf, T);
                daf_copy_ex_row_kernel<<<Bn, 128, 0, stream>>>(exbuf, cat, 517L*Cc, T);
                // rep = leaky(cat @ Wo + bo), M = T+1
                launch_gemm(stream, cat, 517L*Cc, Cc, 1,
                            WB_o, 0, 1, 128,
                            rep, 517L*Cc, Cc, 1,
                            WB_o + 128*128, T + 1, 128, 128, 1, Bn);
                // pred = leaky(rep @ Wdec + bdec), written channel-major (B,129,516)
                launch_gemm(stream, rep, 517L*Cc, Cc, 1,
                            WBd, 0, 1, 128,
                            outb, (long)Dm*TMAX, 1, (long)TMAX,
                            WBd + 129*128, T + 1, 129, 128, 1, Bn);
            }
        }
    }
    #undef FI
}